// ResidualVectorQuantizer_24223615550494
// MI455X (gfx1250) — compile-verified
//
#include <hip/hip_runtime.h>
#include <math.h>

// ---------------- problem constants ----------------
constexpr int   N_TOK   = 32768;
constexpr int   KCB     = 1024;   // codebook entries
constexpr int   DIM     = 128;    // embedding dim
constexpr int   NQ      = 4;      // residual quantizer stages
constexpr int   SK_IT   = 100;    // sinkhorn iterations
constexpr float SK_EPS  = 0.003f;
constexpr float BETA_C  = 0.25f;
constexpr float LN_K    = 6.93147180559945f;   // ln(1024)
constexpr float LN_B    = 10.3972077083992f;   // ln(32768)
constexpr float LOSS_SCALE = (1.0f + BETA_C) / ((float)N_TOK * (float)DIM * (float)NQ);

typedef __attribute__((ext_vector_type(2))) float v2f;
typedef __attribute__((ext_vector_type(8))) float v8f;

// ---------------- init: residual = x, xq = 0, loss = 0 ----------------
__global__ void init_kernel(const float* __restrict__ x, float* __restrict__ residual,
                            float* __restrict__ xq_out, float* __restrict__ lossAcc) {
  int i = blockIdx.x * 256 + threadIdx.x;
  if (i < N_TOK * DIM) {
    residual[i] = x[i];
    xq_out[i]   = 0.0f;
  }
  if (i == 0) *lossAcc = 0.0f;
}

// ---------------- squared row norms (one wave per row) ----------------
__global__ void rownorm_kernel(const float* __restrict__ M, float* __restrict__ out, int nrows) {
  int wid  = threadIdx.x >> 5;
  int lane = threadIdx.x & 31;
  int row  = blockIdx.x * 8 + wid;
  if (row >= nrows) return;
  const float* p = M + (size_t)row * DIM;
  float s = 0.0f;
#pragma unroll
  for (int k = lane; k < DIM; k += 32) { float v = p[k]; s += v * v; }
#pragma unroll
  for (int m = 16; m >= 1; m >>= 1) s += __shfl_xor(s, m, 32);
  if (lane == 0) out[row] = s;
}

// ---------------- WMMA distance GEMM: d = xn + cn - 2 * X * C^T ----------------
// One wave computes one 16x16 tile of d.  A = 16x4 slab of X, B = 4x16 slab of C^T.
// A(m,k): lane = m + 16*(k>=2), vgpr = k&1.  B(k,n): lane = n + 16*(k>=2), vgpr = k&1.
__global__ void gemm_dist_kernel(const float* __restrict__ X, const float* __restrict__ C,
                                 const float* __restrict__ xn, const float* __restrict__ cn,
                                 float* __restrict__ d) {
  int wid  = threadIdx.x >> 5;
  int lane = threadIdx.x & 31;
  int gw   = blockIdx.x * 8 + wid;
  int tn   = gw & 63;          // 1024/16 = 64 tiles along codebook dim
  int tm   = gw >> 6;          // 32768/16 = 2048 tiles along token dim
  int i0 = tm * 16, j0 = tn * 16;
  int r = lane & 15, g = lane >> 4;
  const float* xrow = X + (size_t)(i0 + r) * DIM + 2 * g;
  const float* crow = C + (size_t)(j0 + r) * DIM + 2 * g;
  v8f acc = {};
#pragma unroll
  for (int kk = 0; kk < DIM; kk += 4) {
    v2f a = *(const v2f*)(xrow + kk);
    v2f b = *(const v2f*)(crow + kk);
    acc = __builtin_amdgcn_wmma_f32_16x16x4_f32(false, a, false, b, (short)0, acc, false, false);
  }
  // C/D layout: element (m,n) -> vgpr = m&7, lane = n + 16*(m>>3)
  float cnv = cn[j0 + r];
#pragma unroll
  for (int v = 0; v < 8; ++v) {
    int m = v + 8 * g;
    float val = xn[i0 + m] + cnv - 2.0f * acc[v];
    d[(size_t)(i0 + m) * KCB + j0 + r] = val;
  }
}

// ---------------- global min/max of d (two stage) ----------------
__global__ void minmax_partial(const float* __restrict__ d, float* __restrict__ pmin,
                               float* __restrict__ pmax) {
  __shared__ float smin[256], smax[256];
  float lo = 3.4e38f, hi = -3.4e38f;
  for (size_t i = (size_t)blockIdx.x * 256 + threadIdx.x; i < (size_t)N_TOK * KCB;
       i += (size_t)1024 * 256) {
    float v = d[i];
    lo = fminf(lo, v);
    hi = fmaxf(hi, v);
  }
  smin[threadIdx.x] = lo; smax[threadIdx.x] = hi;
  __syncthreads();
  for (int s = 128; s >= 1; s >>= 1) {
    if (threadIdx.x < s) {
      smin[threadIdx.x] = fminf(smin[threadIdx.x], smin[threadIdx.x + s]);
      smax[threadIdx.x] = fmaxf(smax[threadIdx.x], smax[threadIdx.x + s]);
    }
    __syncthreads();
  }
  if (threadIdx.x == 0) { pmin[blockIdx.x] = smin[0]; pmax[blockIdx.x] = smax[0]; }
}

// scal: [0]=s1, [1]=s0, [2]=amax, [3]=sumExp accumulator
__global__ void minmax_final(const float* __restrict__ pmin, const float* __restrict__ pmax,
                             float* __restrict__ scal) {
  __shared__ float smin[256], smax[256];
  float lo = 3.4e38f, hi = -3.4e38f;
  for (int i = threadIdx.x; i < 1024; i += 256) {
    lo = fminf(lo, pmin[i]);
    hi = fmaxf(hi, pmax[i]);
  }
  smin[threadIdx.x] = lo; smax[threadIdx.x] = hi;
  __syncthreads();
  for (int s = 128; s >= 1; s >>= 1) {
    if (threadIdx.x < s) {
      smin[threadIdx.x] = fminf(smin[threadIdx.x], smin[threadIdx.x + s]);
      smax[threadIdx.x] = fmaxf(smax[threadIdx.x], smax[threadIdx.x + s]);
    }
    __syncthreads();
  }
  if (threadIdx.x == 0) {
    float dmin = smin[0], dmax = smax[0];
    float mid = 0.5f * (dmax + dmin);
    float amp = dmax - mid + 1e-5f;
    float s1  = -1.0f / (amp * SK_EPS);         // a_ij = s1*d_ij + s0
    float s0  = mid / (amp * SK_EPS);
    scal[0] = s1;
    scal[1] = s0;
    scal[2] = s1 * dmin + s0;                   // a_max (s1 < 0)
    scal[3] = 0.0f;                             // sumExp accumulator
  }
}

// ---------------- sum of exp(a - amax) for initial u = 1/S ----------------
__global__ void sumexp_kernel(const float* __restrict__ d, float* __restrict__ scal) {
  __shared__ float sred[256];
  float s1 = scal[0], b = scal[1] - scal[2];
  float acc = 0.0f;
  for (size_t i = (size_t)blockIdx.x * 256 + threadIdx.x; i < (size_t)N_TOK * KCB;
       i += (size_t)1024 * 256)
    acc += __expf(fmaf(d[i], s1, b));
  sred[threadIdx.x] = acc;
  __syncthreads();
  for (int s = 128; s >= 1; s >>= 1) {
    if (threadIdx.x < s) sred[threadIdx.x] += sred[threadIdx.x + s];
    __syncthreads();
  }
  if (threadIdx.x == 0) atomicAdd(&scal[3], sred[0]);
}

// lu_i = -ln(S) = -(amax + ln(sumExp))
__global__ void init_lu_kernel(const float* __restrict__ scal, float* __restrict__ lu) {
  int i = blockIdx.x * 256 + threadIdx.x;
  if (i < N_TOK) lu[i] = -(scal[2] + __logf(scal[3]));
}

__global__ void zero_colsum(float* __restrict__ colsum) {
  int j = blockIdx.x * 256 + threadIdx.x;
  if (j < KCB) colsum[j] = 0.0f;
}

// colsum_j += sum_i exp(s1*d_ij + s0 + lu_i) over a 512-row chunk
__global__ void colsum_kernel(const float* __restrict__ d, const float* __restrict__ lu,
                              const float* __restrict__ scal, float* __restrict__ colsum) {
  int j  = blockIdx.x * 256 + threadIdx.x;
  int r0 = blockIdx.y * 512;
  float s1 = scal[0], s0 = scal[1];
  float acc = 0.0f;
  for (int i = r0; i < r0 + 512; ++i)
    acc += __expf(fmaf(d[(size_t)i * KCB + j], s1, s0 + lu[i]));
  atomicAdd(&colsum[j], acc);
}

// lv_j = -(ln colsum_j + ln K)
__global__ void lv_kernel(const float* __restrict__ colsum, float* __restrict__ lv) {
  int j = blockIdx.x * 256 + threadIdx.x;
  if (j < KCB) lv[j] = -(__logf(colsum[j]) + LN_K);
}

// lu_i = -(ln(sum_j exp(s1*d_ij + s0 + lv_j)) + ln B);  one wave per row
__global__ void rowsum_kernel(const float* __restrict__ d, const float* __restrict__ lv,
                              const float* __restrict__ scal, float* __restrict__ lu) {
  __shared__ float lvs[KCB];
  for (int j = threadIdx.x; j < KCB; j += 256) lvs[j] = lv[j];
  __syncthreads();
  int wid  = threadIdx.x >> 5;
  int lane = threadIdx.x & 31;
  int row  = blockIdx.x * 8 + wid;
  const float* dp = d + (size_t)row * KCB;
  float s1 = scal[0], s0 = scal[1];
  float acc = 0.0f;
#pragma unroll 4
  for (int j = lane; j < KCB; j += 32)
    acc += __expf(fmaf(dp[j], s1, s0 + lvs[j]));
#pragma unroll
  for (int m = 16; m >= 1; m >>= 1) acc += __shfl_xor(acc, m, 32);
  if (lane == 0) lu[row] = -(__logf(acc) + LN_B);
}

// argmax_j (s1*d_ij + lv_j); one wave per row; ties -> lowest j (matches jnp.argmax)
__global__ void argmax_kernel(const float* __restrict__ d, const float* __restrict__ lv,
                              const float* __restrict__ scal, int* __restrict__ idx_out,
                              float* __restrict__ dout_idx, int q) {
  __shared__ float lvs[KCB];
  for (int j = threadIdx.x; j < KCB; j += 256) lvs[j] = lv[j];
  __syncthreads();
  int wid  = threadIdx.x >> 5;
  int lane = threadIdx.x & 31;
  int row  = blockIdx.x * 8 + wid;
  const float* dp = d + (size_t)row * KCB;
  float s1 = scal[0];
  float best = -3.4e38f;
  int   bi   = 0;
  for (int j = lane; j < KCB; j += 32) {
    float v = fmaf(dp[j], s1, lvs[j]);
    if (v > best) { best = v; bi = j; }
  }
#pragma unroll
  for (int m = 16; m >= 1; m >>= 1) {
    float ov = __shfl_xor(best, m, 32);
    int   oi = __shfl_xor(bi, m, 32);
    if (ov > best || (ov == best && oi < bi)) { best = ov; bi = oi; }
  }
  if (lane == 0) {
    idx_out[row] = bi;
    dout_idx[(size_t)row * NQ + q] = (float)bi;
  }
}

// gather codebook row, accumulate xq, update residual, accumulate loss
__global__ void update_kernel(const float* __restrict__ Cb, const int* __restrict__ idx,
                              float* __restrict__ residual, float* __restrict__ xq_out,
                              float* __restrict__ lossAcc) {
  __shared__ float sred[128];
  int row = blockIdx.x;
  int t   = threadIdx.x;
  int id  = idx[row];
  float c = Cb[(size_t)id * DIM + t];
  size_t o = (size_t)row * DIM + t;
  float r = residual[o];
  float diff = c - r;
  xq_out[o] += c;
  residual[o] = r - c;
  sred[t] = diff * diff;
  __syncthreads();
  for (int s = 64; s >= 1; s >>= 1) {
    if (t < s) sred[t] += sred[t + s];
    __syncthreads();
  }
  if (t == 0) atomicAdd(lossAcc, sred[0] * LOSS_SCALE);
}

__global__ void finalize_kernel(const float* __restrict__ lossAcc, float* __restrict__ loss_out) {
  if (threadIdx.x == 0) *loss_out = *lossAcc;
}

// ---------------- launch ----------------
extern "C" void kernel_launch(void* const* d_in, const int* in_sizes, int n_in,
                              void* d_out, int out_size, void* d_ws, size_t ws_size,
                              hipStream_t stream) {
  (void)in_sizes; (void)n_in; (void)out_size; (void)ws_size;
  const float* x         = (const float*)d_in[0];
  const float* codebooks = (const float*)d_in[1];

  float* out       = (float*)d_out;
  float* xq_out    = out;                        // N*D
  float* loss_out  = out + (size_t)N_TOK * DIM;  // 1
  float* idx_out_f = loss_out + 1;               // N*NQ

  float* dmat     = (float*)d_ws;                        // N*K f32
  float* residual = dmat + (size_t)N_TOK * KCB;          // N*D
  float* lu       = residual + (size_t)N_TOK * DIM;      // N
  float* lv       = lu + N_TOK;                          // K
  float* colsum   = lv + KCB;                            // K
  float* pmin     = colsum + KCB;                        // 1024
  float* pmax     = pmin + 1024;                         // 1024
  float* cnorm    = pmax + 1024;                         // K
  float* xnorm    = cnorm + KCB;                         // N
  float* scal     = xnorm + N_TOK;                       // 8
  float* lossAcc  = scal + 8;                            // 8
  int*   idxbuf   = (int*)(lossAcc + 8);                 // N ints

  init_kernel<<<(N_TOK * DIM) / 256, 256, 0, stream>>>(x, residual, xq_out, lossAcc);

  for (int q = 0; q < NQ; ++q) {
    const float* Cb = codebooks + (size_t)q * KCB * DIM;

    rownorm_kernel<<<KCB / 8, 256, 0, stream>>>(Cb, cnorm, KCB);
    rownorm_kernel<<<N_TOK / 8, 256, 0, stream>>>(residual, xnorm, N_TOK);

    gemm_dist_kernel<<<(2048 * 64) / 8, 256, 0, stream>>>(residual, Cb, xnorm, cnorm, dmat);

    minmax_partial<<<1024, 256, 0, stream>>>(dmat, pmin, pmax);
    minmax_final<<<1, 256, 0, stream>>>(pmin, pmax, scal);
    sumexp_kernel<<<1024, 256, 0, stream>>>(dmat, scal);
    init_lu_kernel<<<N_TOK / 256, 256, 0, stream>>>(scal, lu);

    for (int it = 0; it < SK_IT; ++it) {
      zero_colsum<<<KCB / 256, 256, 0, stream>>>(colsum);
      colsum_kernel<<<dim3(KCB / 256, N_TOK / 512), 256, 0, stream>>>(dmat, lu, scal, colsum);
      lv_kernel<<<KCB / 256, 256, 0, stream>>>(colsum, lv);
      rowsum_kernel<<<N_TOK / 8, 256, 0, stream>>>(dmat, lv, scal, lu);
    }

    argmax_kernel<<<N_TOK / 8, 256, 0, stream>>>(dmat, lv, scal, idxbuf, idx_out_f, q);
    update_kernel<<<N_TOK, 128, 0, stream>>>(Cb, idxbuf, residual, xq_out, lossAcc);
  }

  finalize_kernel<<<1, 1, 0, stream>>>(lossAcc, loss_out);
}